// ConsensusField3D_78013785964619
// MI455X (gfx1250) — compile-verified
//
#include <hip/hip_runtime.h>
#include <math.h>

// ConsensusField3D: per-voxel quaternion slerp + gate + quat->R. Memory-bound
// (~212 MB @ 23.3 TB/s ~ 9 us floor) with VALU as co-bottleneck; this version
// moves all reciprocals/sqrts to the trans pipe (v_rcp_f32 / v_sqrt_f32) so
// they co-execute with VALU. Data path: NT B128 loads/stores, L2 prefetch.

typedef float v4f __attribute__((ext_vector_type(4)));

static constexpr int DHW   = 96 * 96 * 96;        // 884736 voxels per batch
static constexpr int NV4   = DHW / 4;             // 221184 float4 groups per batch
static constexpr int GTOT  = 2 * NV4;             // 442368 groups total (B = 2)
static constexpr int U_OFF = 2 * 9 * DHW;         // U_bar region size (floats)
static constexpr int R_OFF = U_OFF + 2 * 3 * DHW; // start of q_out region

#define EPSQ 1e-6f

__device__ __forceinline__ v4f ldnt(const float* p) {
    return __builtin_nontemporal_load(reinterpret_cast<const v4f*>(p));
}
__device__ __forceinline__ void stnt(float* p, v4f v) {
    __builtin_nontemporal_store(v, reinterpret_cast<v4f*>(p));
}
// Raw trans-pipe ops (~1 ulp): avoid IEEE div/sqrt Newton-refinement fixups.
__device__ __forceinline__ float frcp(float x)  { return __builtin_amdgcn_rcpf(x); }
__device__ __forceinline__ float fsqrt(float x) { return __builtin_amdgcn_sqrtf(x); }

__global__ void __launch_bounds__(256)
ConsensusField3D_kernel(const float* __restrict__ r,
                        const float* __restrict__ q,
                        const float* __restrict__ r_ref,
                        const float* __restrict__ q_ref,
                        const float* __restrict__ gate_w,
                        const float* __restrict__ gate_b,
                        float* __restrict__ out)
{
    // Uniform addresses -> scalar (s_load) path, broadcast to the wave.
    const float gw0 = gate_w[0], gw1 = gate_w[1], gw2 = gate_w[2], gw3 = gate_w[3];
    const float gb  = gate_b[0];

    const int stride = gridDim.x * blockDim.x;
    for (int g = blockIdx.x * blockDim.x + threadIdx.x; g < GTOT; g += stride) {
        // Warm L2 one grid-stride tile ahead (global_prefetch_b8).
        const int gn = g + stride;
        if (gn < GTOT) {
            const int bn = gn / NV4;
            const int vn = (gn - bn * NV4) * 4;
            __builtin_prefetch(q     + bn * 4 * DHW + vn, 0, 1);
            __builtin_prefetch(q_ref + bn * 4 * DHW + vn, 0, 1);
            __builtin_prefetch(r     + bn * 3 * DHW + vn, 0, 1);
            __builtin_prefetch(r_ref + bn * 3 * DHW + vn, 0, 1);
        }

        const int b  = g / NV4;
        const int v  = (g - b * NV4) * 4;
        const int qb = b * 4 * DHW + v;
        const int rb = b * 3 * DHW + v;

        // 14x coalesced non-temporal global_load_b128
        v4f qa0 = ldnt(q + qb);               v4f qa1 = ldnt(q + qb + DHW);
        v4f qa2 = ldnt(q + qb + 2 * DHW);     v4f qa3 = ldnt(q + qb + 3 * DHW);
        v4f qr0 = ldnt(q_ref + qb);           v4f qr1 = ldnt(q_ref + qb + DHW);
        v4f qr2 = ldnt(q_ref + qb + 2 * DHW); v4f qr3 = ldnt(q_ref + qb + 3 * DHW);
        v4f ra0 = ldnt(r + rb);               v4f ra1 = ldnt(r + rb + DHW);
        v4f ra2 = ldnt(r + rb + 2 * DHW);
        v4f rr0 = ldnt(r_ref + rb);           v4f rr1 = ldnt(r_ref + rb + DHW);
        v4f rr2 = ldnt(r_ref + rb + 2 * DHW);

        v4f U[9], RB[3], QO[4];

        #pragma unroll
        for (int l = 0; l < 4; ++l) {
            float a0 = qa0[l], a1 = qa1[l], a2 = qa2[l], a3 = qa3[l];
            float b0 = qr0[l], b1 = qr1[l], b2 = qr2[l], b3 = qr3[l];

            // inverse norms (normalization folded into slerp weights below)
            float ina = frcp(fsqrt(a0*a0 + a1*a1 + a2*a2 + a3*a3) + EPSQ);
            float inb = frcp(fsqrt(b0*b0 + b1*b1 + b2*b2 + b3*b3) + EPSQ);

            // dot of normalized quats
            float d = (a0*b0 + a1*b1 + a2*b2 + a3*b3) * (ina * inb);
            float c = fminf(fabsf(d), 1.0f);

            // gate: feat = [r_mean, r_mean_ref, c, 1]
            float rm  = (ra0[l] + ra1[l] + ra2[l]) * (1.0f / 3.0f);
            float rmr = (rr0[l] + rr1[l] + rr2[l]) * (1.0f / 3.0f);
            float zg  = gw0 * rm + gw1 * rmr + gw2 * c + gw3 + gb;
            float omega = frcp(1.0f + __expf(-zg));
            float t = omega;

            // slerp (reference convention: sin(t*th)*q0 + sin((1-t)*th)*q1')
            float sgn = d < 0.0f ? -1.0f : 1.0f;
            float dd  = fminf(fabsf(d), 1.0f - 1e-7f);
            float th  = acosf(dd);
            float st  = fsqrt(fmaxf(1.0f - dd * dd, 0.0f)); // sin(acos(dd))
            bool  sm  = st < 1e-4f;
            float isafe = sm ? 1.0f : frcp(st);
            float w0 = sm ? t          : __sinf(t * th) * isafe;
            float w1 = sm ? (1.0f - t) : __sinf((1.0f - t) * th) * isafe;
            // fold quaternion normalizations + sign flip into the weights
            float W0 = w0 * ina;
            float W1 = w1 * inb * sgn;

            float o0 = W0*a0 + W1*b0, o1 = W0*a1 + W1*b1;
            float o2 = W0*a2 + W1*b2, o3 = W0*a3 + W1*b3;
            float ino = frcp(fsqrt(o0*o0 + o1*o1 + o2*o2 + o3*o3) + EPSQ);
            float qw = o0*ino, qx = o1*ino, qy = o2*ino, qz = o3*ino;

            // quaternion -> rotation matrix
            float xx = qx*qx, yy = qy*qy, zz = qz*qz;
            float xy = qx*qy, xz = qx*qz, yz = qy*qz;
            float wx = qw*qx, wy = qw*qy, wz = qw*qz;
            float m00 = 1.0f - 2.0f*(yy+zz), m01 = 2.0f*(xy-wz),       m02 = 2.0f*(xz+wy);
            float m10 = 2.0f*(xy+wz),       m11 = 1.0f - 2.0f*(xx+zz), m12 = 2.0f*(yz-wx);
            float m20 = 2.0f*(xz-wy),       m21 = 2.0f*(yz+wx),        m22 = 1.0f - 2.0f*(xx+yy);

            // column-normalize (norm over rows = axis -2)
            float ic0 = frcp(fsqrt(m00*m00 + m10*m10 + m20*m20) + EPSQ);
            float ic1 = frcp(fsqrt(m01*m01 + m11*m11 + m21*m21) + EPSQ);
            float ic2 = frcp(fsqrt(m02*m02 + m12*m12 + m22*m22) + EPSQ);

            // U_bar[b, col j, row i, ...] = Rn[i][j]
            U[0][l] = m00*ic0; U[1][l] = m10*ic0; U[2][l] = m20*ic0;
            U[3][l] = m01*ic1; U[4][l] = m11*ic1; U[5][l] = m21*ic1;
            U[6][l] = m02*ic2; U[7][l] = m12*ic2; U[8][l] = m22*ic2;

            RB[0][l] = omega*ra0[l] + (1.0f-omega)*rr0[l];
            RB[1][l] = omega*ra1[l] + (1.0f-omega)*rr1[l];
            RB[2][l] = omega*ra2[l] + (1.0f-omega)*rr2[l];

            QO[0][l] = qw; QO[1][l] = qx; QO[2][l] = qy; QO[3][l] = qz;
        }

        // 16x coalesced non-temporal global_store_b128
        float* up = out + b * 9 * DHW + v;
        #pragma unroll
        for (int j = 0; j < 9; ++j) stnt(up + j * DHW, U[j]);
        float* rp = out + U_OFF + rb;
        #pragma unroll
        for (int k = 0; k < 3; ++k) stnt(rp + k * DHW, RB[k]);
        float* qp = out + R_OFF + qb;
        #pragma unroll
        for (int cc = 0; cc < 4; ++cc) stnt(qp + cc * DHW, QO[cc]);
    }
}

extern "C" void kernel_launch(void* const* d_in, const int* in_sizes, int n_in,
                              void* d_out, int out_size, void* d_ws, size_t ws_size,
                              hipStream_t stream) {
    (void)in_sizes; (void)n_in; (void)out_size; (void)d_ws; (void)ws_size;
    // setup_inputs order: U, r, q, U_ref, r_ref, q_ref, gate_w, gate_b
    // (U and U_ref are never read by the reference computation.)
    const float* r      = (const float*)d_in[1];
    const float* q      = (const float*)d_in[2];
    const float* r_ref  = (const float*)d_in[4];
    const float* q_ref  = (const float*)d_in[5];
    const float* gate_w = (const float*)d_in[6];
    const float* gate_b = (const float*)d_in[7];
    float* out = (float*)d_out;

    dim3 block(256);
    dim3 grid(GTOT / (2 * 256)); // 864 blocks -> exactly 2 grid-stride iters/thread
    hipLaunchKernelGGL(ConsensusField3D_kernel, grid, block, 0, stream,
                       r, q, r_ref, q_ref, gate_w, gate_b, out);
}